// NFTM1DLearnable_4964982194715
// MI455X (gfx1250) — compile-verified
//
#include <hip/hip_runtime.h>
#include <math.h>

// ---------------------------------------------------------------------------
// Heat diffusion scan, store-bandwidth bound (512 MB out @ 23.3 TB/s ~ 22us).
// One workgroup per batch row; whole row double-buffered in LDS (2 x 64 KB,
// fits CDNA5's 320 KB/WGP). Every timestep is written to HBM with the gfx1250
// async LDS->memory DMA path (GLOBAL_STORE_ASYNC_FROM_LDS_B128, ASYNCcnt), so
// stores overlap the next step's stencil compute.
// ---------------------------------------------------------------------------

#define NCOLS   16384          // row length N
#define TPB     1024           // threads per block (32 wave32)
#define GROUPS  4              // 4 x float4 groups per thread
#define GSTRIDE (TPB * 4)      // 4096 elements between a thread's groups

#define ALPHA_MIN_F 0.001f
#define ALPHA_MAX_F 1.0f

// The async-LDS builtins take pointers to 16-byte int vectors:
//   param0: v4i __attribute__((address_space(1)))*   (global)
//   param1: v4i __attribute__((address_space(3)))*   (LDS)
typedef int v4i __attribute__((vector_size(16)));
typedef v4i __attribute__((address_space(1))) *gv4i_ptr;
typedef v4i __attribute__((address_space(3))) *lv4i_ptr;

// ---- CDNA5 async global<->LDS helpers -------------------------------------

__device__ __forceinline__ void async_load_to_lds_b128(const float* gsrc, float* ldst) {
#if __has_builtin(__builtin_amdgcn_global_load_async_to_lds_b128)
  __builtin_amdgcn_global_load_async_to_lds_b128(
      (gv4i_ptr)(unsigned long long)(uintptr_t)gsrc,
      (lv4i_ptr)(unsigned int)(uintptr_t)ldst,
      0, 0);
#else
  unsigned int l = (unsigned int)(uintptr_t)ldst;
  asm volatile("global_load_async_to_lds_b128 %0, %1, off"
               :: "v"(l), "v"(gsrc) : "memory");
#endif
}

__device__ __forceinline__ void async_store_from_lds_b128(float* gdst, const float* lsrc) {
#if __has_builtin(__builtin_amdgcn_global_store_async_from_lds_b128)
  __builtin_amdgcn_global_store_async_from_lds_b128(
      (gv4i_ptr)(unsigned long long)(uintptr_t)gdst,
      (lv4i_ptr)(unsigned int)(uintptr_t)lsrc,
      0, 0);
#else
  unsigned int l = (unsigned int)(uintptr_t)lsrc;
  asm volatile("global_store_async_from_lds_b128 %0, %1, off"
               :: "v"(gdst), "v"(l) : "memory");
#endif
}

#if __has_builtin(__builtin_amdgcn_s_wait_asynccnt)
#define WAIT_ASYNC(n) __builtin_amdgcn_s_wait_asynccnt(n)
#else
#define WAIT_ASYNC(n) asm volatile("s_wait_asynccnt %0" :: "n"(n) : "memory")
#endif

// ---------------------------------------------------------------------------

__global__ __launch_bounds__(TPB, 1) void nftm_heat_rows_kernel(
    const float* __restrict__ f0,
    const float* __restrict__ log_alpha,
    float* __restrict__ out,
    int T)
{
  __shared__ __align__(16) float bufA[NCOLS];   // 64 KB
  __shared__ __align__(16) float bufB[NCOLS];   // 64 KB

  const int b   = blockIdx.x;
  const int tid = threadIdx.x;

  float alpha = expf(log_alpha[0]);
  alpha = fminf(fmaxf(alpha, ALPHA_MIN_F), ALPHA_MAX_F);

  const float* f0row  = f0  + (size_t)b * NCOLS;
  float*       outrow = out + (size_t)b * (size_t)T * NCOLS;

  int e[GROUPS];
#pragma unroll
  for (int g = 0; g < GROUPS; ++g) e[g] = g * GSTRIDE + tid * 4;

  // ---- stage initial row into LDS via async DMA loads ---------------------
#pragma unroll
  for (int g = 0; g < GROUPS; ++g)
    async_load_to_lds_b128(f0row + e[g], &bufA[e[g]]);
  WAIT_ASYNC(0);
  __syncthreads();

  // pull own elements into registers
  float x[GROUPS * 4];
#pragma unroll
  for (int g = 0; g < GROUPS; ++g) {
    float4 v = *(const float4*)&bufA[e[g]];
    x[g*4+0] = v.x; x[g*4+1] = v.y; x[g*4+2] = v.z; x[g*4+3] = v.w;
  }

  // ---- state t = 0 -> out (async LDS->memory) -----------------------------
#pragma unroll
  for (int g = 0; g < GROUPS; ++g)
    async_store_from_lds_b128(outrow + e[g], &bufA[e[g]]);

  float* cur = bufA;
  float* nxt = bufB;

  for (int t = 1; t < T; ++t) {
    // `nxt` was the async-store source 2 steps ago; each step issues 4 async
    // stores per wave, so draining to <=4 releases exactly that buffer while
    // keeping last step's stores in flight.
    WAIT_ASYNC(4);

    float n[GROUPS * 4];
#pragma unroll
    for (int g = 0; g < GROUPS; ++g) {
      const int eg = e[g];
      // edge padding: clamped index reproduces lap[0]=x1-x0, lap[N-1]=x_{N-2}-x_{N-1}
      const float xl = cur[eg > 0 ? eg - 1 : 0];
      const float xr = cur[(eg + 4) < NCOLS ? eg + 4 : (NCOLS - 1)];
      const float a0 = x[g*4+0], a1 = x[g*4+1], a2 = x[g*4+2], a3 = x[g*4+3];
      n[g*4+0] = a0 + alpha * (xl - 2.0f*a0 + a1);
      n[g*4+1] = a1 + alpha * (a0 - 2.0f*a1 + a2);
      n[g*4+2] = a2 + alpha * (a1 - 2.0f*a2 + a3);
      n[g*4+3] = a3 + alpha * (a2 - 2.0f*a3 + xr);
      float4 w; w.x = n[g*4+0]; w.y = n[g*4+1]; w.z = n[g*4+2]; w.w = n[g*4+3];
      *(float4*)&nxt[eg] = w;
    }

    __syncthreads();   // all halo reads of `cur` and writes of `nxt` complete

    float* orow = outrow + (size_t)t * NCOLS;
#pragma unroll
    for (int g = 0; g < GROUPS; ++g)
      async_store_from_lds_b128(orow + e[g], &nxt[e[g]]);

#pragma unroll
    for (int j = 0; j < GROUPS * 4; ++j) x[j] = n[j];
    float* tmp = cur; cur = nxt; nxt = tmp;
  }

  WAIT_ASYNC(0);
}

// ---------------------------------------------------------------------------

extern "C" void kernel_launch(void* const* d_in, const int* in_sizes, int n_in,
                              void* d_out, int out_size, void* d_ws, size_t ws_size,
                              hipStream_t stream) {
  const float* f0        = (const float*)d_in[0];
  const float* log_alpha = (const float*)d_in[1];
  float*       out       = (float*)d_out;

  const int BN = in_sizes[0];         // B * N
  const int B  = BN / NCOLS;          // 128
  const int T  = out_size / BN;       // 64

  nftm_heat_rows_kernel<<<dim3(B), dim3(TPB), 0, stream>>>(f0, log_alpha, out, T);
}